// ErrorBoundedSampler_7928509628627
// MI455X (gfx1250) — compile-verified
//
#include <hip/hip_runtime.h>

// ErrorBoundedSampler PDF resampling for MI455X (gfx1250, wave32).
// One wave32 per ray: coalesced b128 weight load, shuffle-scan CDF,
// async global->LDS staging of existing_bins (ASYNCcnt path), LDS binary
// search + interpolation. Pure bandwidth kernel: ~340 MB -> ~15 us at 23.3 TB/s.

constexpr int   kRays         = 262144;
constexpr int   kEval         = 128;   // NUM_EVAL
constexpr int   kBins         = 65;    // NUM_SAMPLES + 1
constexpr float kHistPad      = 1e-5f;
constexpr float kEps          = 1e-5f;
constexpr int   kRaysPerBlock = 8;     // 8 waves of 32
constexpr int   kRow          = 132;   // 129 used, padded

__global__ __launch_bounds__(256) void ebs_pdf_sample_kernel(
    const float* __restrict__ weights,       // [R,128]
    const float* __restrict__ existing_bins, // [R,129] sorted
    const float* __restrict__ nears,         // [R]
    const float* __restrict__ fars,          // [R]
    float* __restrict__ out)                 // [R,65]
{
    __shared__ float s_cdf [kRaysPerBlock][kRow];
    __shared__ float s_bins[kRaysPerBlock][kRow];

    const int lane = (int)(threadIdx.x & 31u);
    const int wid  = (int)(threadIdx.x >> 5);
    const int ray  = (int)blockIdx.x * kRaysPerBlock + wid;

    // ---- Stage existing_bins row (129 f32) into LDS via CDNA5 async DMA.
    // IOFFSET is added to BOTH the global address and the LDS address, so one
    // base covers elements lane+32k at byte offset 128*k. Lane 0 fetches [128].
    {
        const float* gb = existing_bins + (size_t)ray * (size_t)(kEval + 1);
        unsigned long long ga = (unsigned long long)(uintptr_t)(gb + lane);
        unsigned int       la = (unsigned int)(uintptr_t)(&s_bins[wid][lane]);
        asm volatile(
            "global_load_async_to_lds_b32 %0, %1, off\n\t"
            "global_load_async_to_lds_b32 %0, %1, off offset:128\n\t"
            "global_load_async_to_lds_b32 %0, %1, off offset:256\n\t"
            "global_load_async_to_lds_b32 %0, %1, off offset:384"
            :: "v"(la), "v"(ga) : "memory");
        if (lane == 0) {
            asm volatile(
                "global_load_async_to_lds_b32 %0, %1, off offset:512"
                :: "v"(la), "v"(ga) : "memory");
        }
    }

    // ---- Weights: 4 contiguous floats per lane (coalesced global_load_b128).
    const float4 wv =
        reinterpret_cast<const float4*>(weights + (size_t)ray * kEval)[lane];
    const float w0 = wv.x + kHistPad;
    const float w1 = wv.y + kHistPad;
    const float w2 = wv.z + kHistPad;
    const float w3 = wv.w + kHistPad;
    const float p0 = w0;
    const float p1 = p0 + w1;
    const float p2 = p1 + w2;
    const float p3 = p2 + w3;

    // ---- wave32 inclusive scan of per-lane totals (5 shuffle steps).
    float incl = p3;
    #pragma unroll
    for (int d = 1; d < 32; d <<= 1) {
        float up = __shfl_up(incl, (unsigned)d, 32);
        if (lane >= d) incl += up;
    }
    const float wsum = __shfl(incl, 31, 32);   // total of padded weights
    const float excl = incl - p3;              // exclusive prefix for this lane

    // padding = relu(EPS - wsum); pdf_i = (w_i + padding/N) / (wsum + padding)
    const float padding = fmaxf(0.0f, kEps - wsum);
    const float padEl   = padding * (1.0f / (float)kEval);
    const float invDen  = 1.0f / (wsum + padding);

    // cdf[m] = min(1, (prefix(m) + m*padEl) * invDen), m = 1..128 ; cdf[0] = 0
    const int k0 = 4 * lane + 1;
    s_cdf[wid][k0 + 0] = fminf(1.0f, (excl + p0 + (float)(k0 + 0) * padEl) * invDen);
    s_cdf[wid][k0 + 1] = fminf(1.0f, (excl + p1 + (float)(k0 + 1) * padEl) * invDen);
    s_cdf[wid][k0 + 2] = fminf(1.0f, (excl + p2 + (float)(k0 + 2) * padEl) * invDen);
    s_cdf[wid][k0 + 3] = fminf(1.0f, (excl + p3 + (float)(k0 + 3) * padEl) * invDen);
    if (lane == 0) s_cdf[wid][0] = 0.0f;

    const float nearv = nears[ray];
    const float farv  = fars[ray];

    // Async LDS writes must land before the gather phase (per-wave data only,
    // so no workgroup barrier needed — just drain ASYNCcnt).
    asm volatile("s_wait_asynccnt 0" ::: "memory");

    // ---- 65 samples: lane handles j, j+32 (lane 0 also j=64).
    for (int j = lane; j < kBins; j += 32) {
        const float u = (float)j * (1.0f / (float)kBins)
                      + (1.0f / (2.0f * (float)kBins));

        // searchsorted(cdf, u, side='right') over 129 entries (8 LDS probes)
        int lo = 0, hi = kEval + 1;
        while (lo < hi) {
            const int   mid = (lo + hi) >> 1;
            const float c   = s_cdf[wid][mid];
            if (c <= u) lo = mid + 1; else hi = mid;
        }
        int below = lo - 1;
        below = below < 0 ? 0 : (below > kEval ? kEval : below);
        const int above = lo > kEval ? kEval : lo;

        const float c0 = s_cdf [wid][below];
        const float c1 = s_cdf [wid][above];
        const float b0 = s_bins[wid][below];
        const float b1 = s_bins[wid][above];

        const float dT = c1 - c0;                 // >= 0 (cdf non-decreasing)
        float t = (dT > 0.0f) ? (u - c0) / dT : 0.0f;  // nan/inf -> 0
        t = fminf(fmaxf(t, 0.0f), 1.0f);

        const float b = b0 + t * (b1 - b0);
        out[(size_t)ray * kBins + j] = b * farv + (1.0f - b) * nearv;
    }
}

extern "C" void kernel_launch(void* const* d_in, const int* in_sizes, int n_in,
                              void* d_out, int out_size, void* d_ws, size_t ws_size,
                              hipStream_t stream) {
    (void)in_sizes; (void)n_in; (void)out_size; (void)d_ws; (void)ws_size;
    const float* weights = (const float*)d_in[0];
    const float* ebins   = (const float*)d_in[1];
    const float* nears   = (const float*)d_in[2];
    const float* fars    = (const float*)d_in[3];
    float* out = (float*)d_out;

    dim3 grid(kRays / kRaysPerBlock);   // 32768 blocks
    dim3 block(256);                    // 8 wave32 = 8 rays / block
    ebs_pdf_sample_kernel<<<grid, block, 0, stream>>>(weights, ebins, nears, fars, out);
}